// AWTConv2d_63239098466958
// MI455X (gfx1250) — compile-verified
//
#include <hip/hip_runtime.h>
#include <math.h>

// ---------------------------------------------------------------------------
// AWTConv2d forward for MI455X (gfx1250).  fp32 throughout; final 1x1 conv
// (op_w, C x C per pixel over B*H*W pixels) runs on V_WMMA_F32_16X16X4_F32
// with full-N register blocking (6 accumulators) so out_pre streams from HBM
// exactly once.  Everything else is bandwidth-bound and fused into minimal
// passes.
// ---------------------------------------------------------------------------

#define DEVFN __device__ __forceinline__

namespace {
constexpr int  B    = 8;
constexpr int  C    = 96;
constexpr int  Hf   = 192;
constexpr int  Wf   = 192;
constexpr int  HIDN = 32;
constexpr int  MIDN = 12;          // max(C/8, 4)
constexpr int  C4   = 4 * C;       // 384
constexpr int  H1   = 96,  W1 = 96;
constexpr int  H2v  = 48,  W2v = 48;
constexpr long HWf  = (long)Hf * Wf;     // 36864
constexpr long HW1  = (long)H1 * W1;     // 9216
constexpr long HW2  = (long)H2v * W2v;   // 2304
constexpr long N0   = (long)B * C4 * HW1;   // level-0 4C tensor elems
constexpr long N1   = (long)B * C4 * HW2;   // level-1 4C tensor elems
constexpr long G0N  = (long)C4 * HW1;       // batch-independent gate map L0
constexpr long G1N  = (long)C4 * HW2;
constexpr long NS   = (long)B * HWf;        // one full-res single-channel map
constexpr int  NTILES = C / 16;             // 6 N-tiles in the final GEMM
}

DEVFN int refl(int i, int n) { if (i < 0) i = -i; if (i >= n) i = 2 * n - 2 - i; return i; }
DEVFN float sigm(float x)    { return 1.0f / (1.0f + expf(-x)); }

typedef __attribute__((ext_vector_type(2))) float v2f;
typedef __attribute__((ext_vector_type(8))) float v8f;

// ---------------------------------------------------------------------------
// Forward Haar DWT: src (B,C,h,w) view -> dst (B,4C,h/2,w/2) contiguous.
// sB/sC allow a strided channel view (level-1 reads raw LL band of level-0).
// ---------------------------------------------------------------------------
__global__ void k_wt(const float* __restrict__ src, long sB, long sC,
                     int h, int w, const float* __restrict__ filt,
                     float* __restrict__ dst, long n)
{
    long idx = (long)blockIdx.x * blockDim.x + threadIdx.x;
    if (idx >= n) return;
    int w2 = w >> 1, h2 = h >> 1;
    int x2 = (int)(idx % w2); long t = idx / w2;
    int y2 = (int)(t % h2);  t /= h2;
    int c  = (int)(t % C);   int b = (int)(t / C);
    const float* sp = src + (long)b * sB + (long)c * sC + (long)(2 * y2) * w + 2 * x2;
    float v00 = sp[0], v01 = sp[1], v10 = sp[w], v11 = sp[w + 1];
    long hw2 = (long)h2 * w2;
    float* dp = dst + ((long)b * C4 + c * 4) * hw2 + (long)y2 * w2 + x2;
    const float* f = filt + c * 16;
#pragma unroll
    for (int i = 0; i < 4; ++i)
        dp[(long)i * hw2] = v00 * f[i * 4 + 0] + v01 * f[i * 4 + 1] +
                            v10 * f[i * 4 + 2] + v11 * f[i * 4 + 3];
}

// ---------------------------------------------------------------------------
// Depthwise 5x5, zero-padded SAME; optional bias and per-channel scale (base).
// ---------------------------------------------------------------------------
__global__ void k_dwconv5(const float* __restrict__ src, float* __restrict__ dst,
                          const float* __restrict__ wgt, const float* __restrict__ bias,
                          const float* __restrict__ scale, int CH, int h, int w, long n)
{
    long idx = (long)blockIdx.x * blockDim.x + threadIdx.x;
    if (idx >= n) return;
    int x = (int)(idx % w); long t = idx / w;
    int y = (int)(t % h);  t /= h;
    int ch = (int)(t % CH); int b = (int)(t / CH);
    const float* sp = src + ((long)b * CH + ch) * (long)h * w;
    const float* wp = wgt + (long)ch * 25;
    float acc = 0.f;
#pragma unroll
    for (int r = 0; r < 5; ++r) {
        int yy = y + r - 2; if (yy < 0 || yy >= h) continue;
#pragma unroll
        for (int s = 0; s < 5; ++s) {
            int xx = x + s - 2; if (xx < 0 || xx >= w) continue;
            acc += wp[r * 5 + s] * sp[(long)yy * w + xx];
        }
    }
    if (bias)  acc += bias[ch];
    if (scale) acc *= scale[ch];
    dst[idx] = acc;
}

// ---------------------------------------------------------------------------
// Coord-gate map (batch independent): gate[c4, y, x]. One block per pixel,
// 384 threads; hidden layer (32) in LDS.
// ---------------------------------------------------------------------------
__global__ void k_gatemap(const float* __restrict__ w1, const float* __restrict__ b1,
                          const float* __restrict__ w2, const float* __restrict__ b2,
                          float* __restrict__ G, int h, int w)
{
    __shared__ float hdn[HIDN];
    long pix = blockIdx.x;
    int y = (int)(pix / w), x = (int)(pix % w);
    if (threadIdx.x < HIDN) {
        float gx = -1.f + 2.f * (float)x / (float)(w - 1);
        float gy = -1.f + 2.f * (float)y / (float)(h - 1);
        int j = threadIdx.x;
        float v = w1[j * 2] * gx + w1[j * 2 + 1] * gy + b1[j];
        hdn[j] = v > 0.f ? v : 0.f;
    }
    __syncthreads();
    int c = threadIdx.x;                 // 0..4C-1
    float acc = b2[c];
#pragma unroll
    for (int j = 0; j < HIDN; ++j) acc += w2[c * HIDN + j] * hdn[j];
    G[(long)c * h * w + pix] = sigm(acc);
}

// ---------------------------------------------------------------------------
// Fused: per-channel 4x4 subband mix (+bias), then multiply by coord gate.
// One thread handles all 4 subbands of one (b, c, pixel).
// ---------------------------------------------------------------------------
__global__ void k_mix_gate(const float* __restrict__ src, float* __restrict__ dst,
                           const float* __restrict__ mw, const float* __restrict__ mb,
                           const float* __restrict__ G, long hw, long n)
{
    long idx = (long)blockIdx.x * blockDim.x + threadIdx.x;
    if (idx >= n) return;
    long pix = idx % hw; long t = idx / hw;
    int c = (int)(t % C); int b = (int)(t / C);
    const float* sp = src + ((long)b * C4 + c * 4) * hw + pix;
    float t0 = sp[0], t1 = sp[hw], t2 = sp[2 * hw], t3 = sp[3 * hw];
    float* dp = dst + ((long)b * C4 + c * 4) * hw + pix;
    const float* gw = G + (long)c * 4 * hw + pix;
#pragma unroll
    for (int o = 0; o < 4; ++o) {
        const float* w = mw + c * 16 + o * 4;
        float v = w[0] * t0 + w[1] * t1 + w[2] * t2 + w[3] * t3 + mb[c * 4 + o];
        dp[(long)o * hw] = v * gw[(long)o * hw];
    }
}

// ---------------------------------------------------------------------------
// Horizontal strip pool (k=7, reflect) + high/low recombination -> ho.
// ---------------------------------------------------------------------------
__global__ void k_striph(const float* __restrict__ src, float* __restrict__ dst,
                         const float* __restrict__ hl, const float* __restrict__ hh,
                         int h, int w, long n)
{
    long idx = (long)blockIdx.x * blockDim.x + threadIdx.x;
    if (idx >= n) return;
    int x = (int)(idx % w); long t = idx / w;
    int y = (int)(t % h);  t /= h;
    int c4 = (int)(t % C4); int b = (int)(t / C4);
    const float* row = src + (((long)b * C4 + c4) * h + y) * (long)w;
    float s = 0.f;
#pragma unroll
    for (int d = -3; d <= 3; ++d) s += row[refl(x + d, w)];
    float low = s * (1.f / 7.f);
    float tv  = row[x];
    dst[idx] = hl[c4] * low + (1.f + hh[c4]) * (tv - low);
}

// ---------------------------------------------------------------------------
// Vertical strip pool over ho + recombination, then t = t*beta + vo*gamma
// (in place on the t buffer; each thread touches only its own t element).
// ---------------------------------------------------------------------------
__global__ void k_stripv(float* __restrict__ tbuf, const float* __restrict__ ho,
                         const float* __restrict__ vl, const float* __restrict__ vh,
                         const float* __restrict__ beta, const float* __restrict__ gamma,
                         int h, int w, long n)
{
    long idx = (long)blockIdx.x * blockDim.x + threadIdx.x;
    if (idx >= n) return;
    int x = (int)(idx % w); long t = idx / w;
    int y = (int)(t % h);  t /= h;
    int c4 = (int)(t % C4); int b = (int)(t / C4);
    const float* col = ho + ((long)b * C4 + c4) * (long)h * w + x;
    float s = 0.f;
#pragma unroll
    for (int d = -3; d <= 3; ++d) s += col[(long)refl(y + d, h) * w];
    float low = s * (1.f / 7.f);
    float hc  = col[(long)y * w];
    float vo  = vl[c4] * low + (1.f + vh[c4]) * (hc - low);
    tbuf[idx] = tbuf[idx] * beta[c4] + vo * gamma[c4];
}

// ---------------------------------------------------------------------------
// Per-(block) spatial mean reduction: one block per (b, channel).
// ---------------------------------------------------------------------------
__global__ void k_poolmean(const float* __restrict__ src, float* __restrict__ dst, long hw)
{
    __shared__ float sm[256];
    long blk = blockIdx.x;
    const float* base = src + blk * hw;
    float s = 0.f;
    for (long i = threadIdx.x; i < hw; i += blockDim.x) s += base[i];
    sm[threadIdx.x] = s;
    __syncthreads();
    for (int off = blockDim.x >> 1; off > 0; off >>= 1) {
        if ((int)threadIdx.x < off) sm[threadIdx.x] += sm[threadIdx.x + off];
        __syncthreads();
    }
    if (threadIdx.x == 0) dst[blk] = sm[0] / (float)hw;
}

// Subband attention: attn[b,c,o] = sigmoid(sum_i sb_w[c,o,i]*pooled[b,c,i]+sb_b)
__global__ void k_sbattn(const float* __restrict__ pooled, const float* __restrict__ w,
                         const float* __restrict__ bb, float* __restrict__ attn, int n)
{
    int tid = blockIdx.x * blockDim.x + threadIdx.x;
    if (tid >= n) return;
    int o = tid & 3; int c = (tid >> 2) % C; int b = tid / (4 * C);
    float acc = bb[c * 4 + o];
#pragma unroll
    for (int i = 0; i < 4; ++i) acc += w[c * 16 + o * 4 + i] * pooled[b * C4 + c * 4 + i];
    attn[tid] = sigm(acc);
}

// t *= attn[b,c4] * wscale[c4]
__global__ void k_scaleattn(float* __restrict__ t, const float* __restrict__ attn,
                            const float* __restrict__ ws, long hw, long n)
{
    long idx = (long)blockIdx.x * blockDim.x + threadIdx.x;
    if (idx >= n) return;
    long q = idx / hw; int c4 = (int)(q % C4); int b = (int)(q / C4);
    t[idx] *= attn[b * C4 + c4] * ws[c4];
}

// ---------------------------------------------------------------------------
// Inverse Haar DWT (lhs_dilation=2, pad 1, 2x2).  Each fine pixel takes one
// coarse pixel per subband: out[y,x] = sum_i ct[i, y>>1, x>>1] *
// iwt[c, i, 1-(y&1), 1-(x&1)].  nxt (finer accumulation) folded into LL.
// ---------------------------------------------------------------------------
__global__ void k_iwt(const float* __restrict__ ct, const float* __restrict__ nxt,
                      const float* __restrict__ filt, float* __restrict__ out,
                      int h2, int w2, long n)
{
    long idx = (long)blockIdx.x * blockDim.x + threadIdx.x;
    if (idx >= n) return;
    int w = 2 * w2, h = 2 * h2;
    int x = (int)(idx % w); long t = idx / w;
    int y = (int)(t % h);  t /= h;
    int c = (int)(t % C);  int b = (int)(t / C);
    int iy = y >> 1, ix = x >> 1;
    int kidx = (1 - (y & 1)) * 2 + (1 - (x & 1));
    long hw2 = (long)h2 * w2;
    const float* cp = ct + ((long)b * C4 + c * 4) * hw2 + (long)iy * w2 + ix;
    float v0 = cp[0];
    if (nxt) v0 += nxt[((long)b * C + c) * hw2 + (long)iy * w2 + ix];
    const float* f = filt + c * 16 + kidx;
    out[idx] = v0 * f[0] + cp[hw2] * f[4] + cp[2 * hw2] * f[8] + cp[3 * hw2] * f[12];
}

// init = base + sigmoid(fuse_gate[c]) * x_tag
__global__ void k_init(const float* __restrict__ a, const float* __restrict__ xt,
                       const float* __restrict__ fg, float* __restrict__ out, long n)
{
    long idx = (long)blockIdx.x * blockDim.x + threadIdx.x;
    if (idx >= n) return;
    int c = (int)((idx / HWf) % C);
    out[idx] = a[idx] + sigm(fg[c]) * xt[idx];
}

// Channel attention MLP (C -> 12 -> C), one block per batch image.
__global__ void k_cattn(const float* __restrict__ pooled, const float* __restrict__ w1,
                        const float* __restrict__ b1, const float* __restrict__ w2,
                        const float* __restrict__ b2, float* __restrict__ cattn)
{
    __shared__ float mid[MIDN];
    int b = blockIdx.x;
    if ((int)threadIdx.x < MIDN) {
        int m = threadIdx.x;
        float acc = b1[m];
        for (int c = 0; c < C; ++c) acc += w1[m * C + c] * pooled[b * C + c];
        mid[m] = acc > 0.f ? acc : 0.f;
    }
    __syncthreads();
    int o = threadIdx.x;
    float acc = b2[o];
#pragma unroll
    for (int m = 0; m < MIDN; ++m) acc += w2[o * MIDN + m] * mid[m];
    cattn[b * C + o] = acc;
}

// Channel mean + max maps of init.
__global__ void k_savgmax(const float* __restrict__ init, float* __restrict__ savg,
                          float* __restrict__ smax, long n)
{
    long idx = (long)blockIdx.x * blockDim.x + threadIdx.x;
    if (idx >= n) return;
    long pix = idx % HWf; int b = (int)(idx / HWf);
    const float* base = init + (long)b * C * HWf + pix;
    float s = 0.f, m = base[0];
#pragma unroll 4
    for (int c = 0; c < C; ++c) { float v = base[(long)c * HWf]; s += v; m = fmaxf(m, v); }
    savg[idx] = s * (1.f / (float)C);
    smax[idx] = m;
}

// Spatial attention: 7x7 reflect conv over [savg, smax].
__global__ void k_sattn(const float* __restrict__ savg, const float* __restrict__ smax,
                        const float* __restrict__ w, const float* __restrict__ bias,
                        float* __restrict__ sattn, long n)
{
    long idx = (long)blockIdx.x * blockDim.x + threadIdx.x;
    if (idx >= n) return;
    long pix = idx % HWf; int b = (int)(idx / HWf);
    int y = (int)(pix / Wf), x = (int)(pix % Wf);
    float acc = bias[0];
    for (int ky = 0; ky < 7; ++ky) {
        int yy = refl(y + ky - 3, Hf);
        for (int kx = 0; kx < 7; ++kx) {
            int xx = refl(x + kx - 3, Wf);
            long p = (long)b * HWf + (long)yy * Wf + xx;
            acc += w[ky * 7 + kx] * savg[p] + w[49 + ky * 7 + kx] * smax[p];
        }
    }
    sattn[idx] = acc;
}

// ---------------------------------------------------------------------------
// Grouped 7x7 pixel-gate conv over concat([init, pattn]) (groups=C), fused
// with out_pre = init + gate*base + (1-gate)*(sigmoid(fg)*x_tag).
// Group c reads concat channels {2c, 2c+1}: <C -> init, >=C -> cattn+sattn.
// ---------------------------------------------------------------------------
__global__ void k_gate_outpre(const float* __restrict__ init, const float* __restrict__ base,
                              const float* __restrict__ xtag, const float* __restrict__ cattn,
                              const float* __restrict__ sattn, const float* __restrict__ pw,
                              const float* __restrict__ pb, const float* __restrict__ fg,
                              float* __restrict__ out, long n)
{
    long idx = (long)blockIdx.x * blockDim.x + threadIdx.x;
    if (idx >= n) return;
    int x = (int)(idx % Wf); long t = idx / Wf;
    int y = (int)(t % Hf);  t /= Hf;
    int c = (int)(t % C);   int b = (int)(t / C);
    float acc = pb[c];
#pragma unroll
    for (int j = 0; j < 2; ++j) {
        int k = 2 * c + j;
        const float* wp = pw + ((long)c * 2 + j) * 49;
        if (k < C) {
            const float* ip = init + ((long)b * C + k) * HWf;
            for (int ky = 0; ky < 7; ++ky) {
                int yy = refl(y + ky - 3, Hf);
                for (int kx = 0; kx < 7; ++kx)
                    acc += wp[ky * 7 + kx] * ip[(long)yy * Wf + refl(x + kx - 3, Wf)];
            }
        } else {
            float cv = cattn[b * C + (k - C)];
            const float* sp = sattn + (long)b * HWf;
            for (int ky = 0; ky < 7; ++ky) {
                int yy = refl(y + ky - 3, Hf);
                for (int kx = 0; kx < 7; ++kx)
                    acc += wp[ky * 7 + kx] * (cv + sp[(long)yy * Wf + refl(x + kx - 3, Wf)]);
            }
        }
    }
    float gate = sigm(acc);
    float a  = base[idx];
    float yv = sigm(fg[c]) * xtag[idx];
    out[idx] = init[idx] + gate * a + (1.f - gate) * yv;
}

// ---------------------------------------------------------------------------
// Final 1x1 conv as WMMA GEMM: out[b,o,p] = sum_c op_w[o,c]*X[b,c,p] + op_b[o]
// M = pixels, N = Cout(96), K = Cin(96).  One wave owns a 16-pixel tile and
// ALL 6 N-tiles (6 x v8f accumulators): the A fragment is loaded once per
// K-step and reused by 6 WMMAs, so X streams from HBM exactly once; op_w
// (36 KB) stays L2-resident.  Task count is an exact multiple of waves/block
// so EXEC stays all-ones inside every executing wave (WMMA requirement).
// A frag: lane m%16 holds K={0,1}(lo half)/{2,3}(hi half) in 2 VGPRs.
// D frag: VGPR r -> M = r + 8*(lane>=16), N = lane%16.
// ---------------------------------------------------------------------------
__global__ void k_opgemm(const float* __restrict__ X, const float* __restrict__ Wt,
                         const float* __restrict__ bias, float* __restrict__ out)
{
    const int  lane = threadIdx.x & 31;
    const int  wave = threadIdx.x >> 5;
    const long task = (long)blockIdx.x * (blockDim.x >> 5) + wave;  // pixel tile
    const long PT   = (long)B * HWf / 16;                           // 18432
    if (task >= PT) return;                                         // wave-uniform
    const int  b    = (int)(task / (HWf / 16));
    const long pix0 = (task % (HWf / 16)) * 16;
    const int  half = lane >> 4;
    const int  lm   = lane & 15;

    v8f acc[NTILES] = {};
    const float* xb = X + (long)b * C * HWf + pix0 + lm;
    for (int k0 = 0; k0 < C; k0 += 4) {
        const int ka = k0 + 2 * half;
        v2f a;
        a.x = xb[(long)ka * HWf];
        a.y = xb[(long)(ka + 1) * HWf];
#pragma unroll
        for (int nt = 0; nt < NTILES; ++nt) {
            const int n = nt * 16 + lm;
            v2f w;
            w.x = Wt[(long)n * C + ka];
            w.y = Wt[(long)n * C + ka + 1];
            acc[nt] = __builtin_amdgcn_wmma_f32_16x16x4_f32(false, a, false, w,
                                                            (short)0, acc[nt], false, false);
        }
    }
#pragma unroll
    for (int nt = 0; nt < NTILES; ++nt) {
        const int   n  = nt * 16 + lm;
        const float bb = bias[n];
        float* ob = out + ((long)b * C + n) * HWf + pix0 + 8 * half;
#pragma unroll
        for (int r = 0; r < 8; ++r) ob[r] = acc[nt][r] + bb;
    }
}

// ---------------------------------------------------------------------------
// Host driver
// ---------------------------------------------------------------------------
static inline unsigned gblk(long n, int bs) { return (unsigned)((n + bs - 1) / bs); }

extern "C" void kernel_launch(void* const* d_in, const int* in_sizes, int n_in,
                              void* d_out, int out_size, void* d_ws, size_t ws_size,
                              hipStream_t stream)
{
    (void)in_sizes; (void)n_in; (void)out_size; (void)ws_size;
    const float* x        = (const float*)d_in[0];
    const float* wt_filt  = (const float*)d_in[1];
    const float* iwt_filt = (const float*)d_in[2];
    const float* base_w   = (const float*)d_in[3];
    const float* base_b   = (const float*)d_in[4];
    const float* base_sc  = (const float*)d_in[5];
    const float* dw_w     = (const float*)d_in[6];
    const float* mix_w    = (const float*)d_in[7];
    const float* mix_b    = (const float*)d_in[8];
    const float* cg_w1    = (const float*)d_in[9];
    const float* cg_b1    = (const float*)d_in[10];
    const float* cg_w2    = (const float*)d_in[11];
    const float* cg_b2    = (const float*)d_in[12];
    const float* hori_lo  = (const float*)d_in[13];
    const float* hori_hi  = (const float*)d_in[14];
    const float* vert_lo  = (const float*)d_in[15];
    const float* vert_hi  = (const float*)d_in[16];
    const float* gamma    = (const float*)d_in[17];
    const float* beta     = (const float*)d_in[18];
    const float* sb_w     = (const float*)d_in[19];
    const float* sb_b     = (const float*)d_in[20];
    const float* wscale   = (const float*)d_in[21];
    const float* fuse_g   = (const float*)d_in[22];
    const float* ca_w1    = (const float*)d_in[23];
    const float* ca_b1    = (const float*)d_in[24];
    const float* ca_w2    = (const float*)d_in[25];
    const float* ca_b2    = (const float*)d_in[26];
    const float* sa_w     = (const float*)d_in[27];
    const float* sa_b     = (const float*)d_in[28];
    const float* pg_w     = (const float*)d_in[29];
    const float* pg_b     = (const float*)d_in[30];
    const float* op_w     = (const float*)d_in[31];
    const float* op_b     = (const float*)d_in[32];
    float* out = (float*)d_out;

    float* ws   = (float*)d_ws;
    float* BIG0 = ws;                 // L0 t  -> later base
    float* BIG1 = BIG0 + N0;          // L0 ho -> later x_tag
    float* BIG2 = BIG1 + N0;          // init
    float* BIG3 = BIG2 + N0;          // out_pre (GEMM input)
    float* A1p  = BIG3 + N0;          // L1 t
    float* B1p  = A1p + N1;           // L1 ho -> later nxt(96x96)
    float* G0p  = B1p + N1;
    float* G1p  = G0p + G0N;
    float* SAVG = G1p + G1N;
    float* SMAX = SAVG + NS;
    float* SATT = SMAX + NS;
    float* POOL4 = SATT + NS;         // B*4C
    float* ATTN4 = POOL4 + (long)B * C4;
    float* POOLC = ATTN4 + (long)B * C4;
    float* CATT  = POOLC + (long)B * C;

    const int TB = 256;

    // ---- forward DWTs (level-1 consumes the RAW LL band of level-0) -------
    k_wt<<<gblk((long)B * C * HW1, TB), TB, 0, stream>>>(
        x, (long)C * HWf, HWf, Hf, Wf, wt_filt, BIG0, (long)B * C * HW1);
    k_wt<<<gblk((long)B * C * HW2, TB), TB, 0, stream>>>(
        BIG0, (long)C4 * HW1, 4 * HW1, H1, W1, wt_filt, A1p, (long)B * C * HW2);

    // ---- per-level processing ---------------------------------------------
    for (int l = 0; l < 2; ++l) {
        float* T = l == 0 ? BIG0 : A1p;
        float* S = l == 0 ? BIG1 : B1p;
        float* G = l == 0 ? G0p  : G1p;
        const int  h = l == 0 ? H1 : H2v, w = l == 0 ? W1 : W2v;
        const long hw = (long)h * w;
        const long n  = (long)B * C4 * hw;

        k_dwconv5<<<gblk(n, TB), TB, 0, stream>>>(
            T, S, dw_w + (long)l * C4 * 25, nullptr, nullptr, C4, h, w, n);
        k_gatemap<<<(unsigned)hw, C4, 0, stream>>>(
            cg_w1 + l * HIDN * 2, cg_b1 + l * HIDN,
            cg_w2 + (long)l * C4 * HIDN, cg_b2 + l * C4, G, h, w);
        k_mix_gate<<<gblk((long)B * C * hw, TB), TB, 0, stream>>>(
            S, T, mix_w + (long)l * C * 16, mix_b + (long)l * C * 4, G, hw,
            (long)B * C * hw);
        k_striph<<<gblk(n, TB), TB, 0, stream>>>(
            T, S, hori_lo + l * C4, hori_hi + l * C4, h, w, n);
        k_stripv<<<gblk(n, TB), TB, 0, stream>>>(
            T, S, vert_lo + l * C4, vert_hi + l * C4, beta + l * C4,
            gamma + l * C4, h, w, n);
        k_poolmean<<<(unsigned)(B * C4), 256, 0, stream>>>(T, POOL4, hw);
        k_sbattn<<<gblk(B * C * 4, 128), 128, 0, stream>>>(
            POOL4, sb_w + (long)l * C * 16, sb_b + (long)l * C * 4, ATTN4, B * C * 4);
        k_scaleattn<<<gblk(n, TB), TB, 0, stream>>>(T, ATTN4, wscale + l * C4, hw, n);
    }

    // ---- inverse DWTs (cll = LL + nxt folded in) --------------------------
    k_iwt<<<gblk((long)B * C * HW1, TB), TB, 0, stream>>>(
        A1p, nullptr, iwt_filt, B1p, H2v, W2v, (long)B * C * HW1);
    k_iwt<<<gblk((long)B * C * HWf, TB), TB, 0, stream>>>(
        BIG0, B1p, iwt_filt, BIG1, H1, W1, (long)B * C * HWf);   // BIG1 = x_tag

    // ---- base branch + CGA fusion -----------------------------------------
    const long nf = (long)B * C * HWf;
    k_dwconv5<<<gblk(nf, TB), TB, 0, stream>>>(
        x, BIG0, base_w, base_b, base_sc, C, Hf, Wf, nf);          // BIG0 = base
    k_init<<<gblk(nf, TB), TB, 0, stream>>>(BIG0, BIG1, fuse_g, BIG2, nf);
    k_poolmean<<<(unsigned)(B * C), 256, 0, stream>>>(BIG2, POOLC, HWf);
    k_cattn<<<B, C, 0, stream>>>(POOLC, ca_w1, ca_b1, ca_w2, ca_b2, CATT);
    k_savgmax<<<gblk(NS, TB), TB, 0, stream>>>(BIG2, SAVG, SMAX, NS);
    k_sattn<<<gblk(NS, TB), TB, 0, stream>>>(SAVG, SMAX, sa_w, sa_b, SATT, NS);
    k_gate_outpre<<<gblk(nf, TB), TB, 0, stream>>>(
        BIG2, BIG0, BIG1, CATT, SATT, pg_w, pg_b, fuse_g, BIG3, nf);

    // ---- final 1x1 conv on WMMA (fp32), N fully register-blocked ----------
    const long ptiles = (long)B * HWf / 16;                        // 18432
    const unsigned gb = (unsigned)(ptiles / (256 / 32));           // 2304 blocks
    k_opgemm<<<gb, 256, 0, stream>>>(BIG3, op_w, op_b, out);
}